// LstmDecoderWithAttention_44581760533065
// MI455X (gfx1250) — compile-verified
//
#include <hip/hip_runtime.h>
#include <hip/hip_bf16.h>

typedef __attribute__((ext_vector_type(2))) float v2f;
typedef __attribute__((ext_vector_type(8))) float v8f;

#define B_    64
#define P_    196
#define S_    20
#define ENC_  512
#define EMB_  512
#define DEC_  512
#define ATT_  512
#define VOCAB_ 10000
#define KCAT_ 1536   // ENC + EMB + DEC
#define NGATE_ 2048  // 4*DEC

// ---------------------------------------------------------------------------
// Generic fp32 WMMA GEMM:  C[M,N] = A[M,K] @ W[N,K]^T + bias, optional ReLU,
// optional mirrored second output (C2) with its own leading dim.
// One wave computes a 32x64 strip of C via V_WMMA_F32_16X16X4_F32:
//   2 M-subtiles x 4 N-subtiles -> 8 v8f accumulators.
// Inner loop processes 16 K-values at a time: 24 independent b64 loads into
// distinct registers, then 32 WMMAs — so waits are amortized/staggered instead
// of one full s_wait_loadcnt 0 per WMMA.
// Fragment layouts (ISA): lane<16 holds K = k0..k0+1, lane>=16 holds K = k0+2..k0+3,
// row = tile + (lane&15); contiguous float2 from row-major [rows,K] storage.
// ---------------------------------------------------------------------------
__global__ __launch_bounds__(256) void wmma_gemm_f32(
    const float* __restrict__ A, const float* __restrict__ W,
    const float* __restrict__ bias, float* __restrict__ C,
    float* __restrict__ C2,
    int M, int N, int K, int ldc, int ldc2, int nstrips, int act)
{
  const int wave = threadIdx.x >> 5;
  const int lane = threadIdx.x & 31;
  const int gid  = blockIdx.x * 8 + wave;
  const int mt   = gid / nstrips;
  const int ns   = gid - mt * nstrips;
  const int tileM = mt * 32;
  if (tileM >= M) return;                    // wave-uniform: EXEC stays all-ones
  const int tileN = ns * 64;
  const int lm = lane & 15;
  const int kk = (lane >> 4) * 2;

  const float* __restrict__ arow0 = A + (size_t)(tileM + lm) * K + kk;
  const float* __restrict__ arow1 = A + (size_t)(tileM + 16 + lm) * K + kk;
  const float* brow[4];
  int ncol[4];
#pragma unroll
  for (int j = 0; j < 4; ++j) {
    int n = tileN + j * 16 + lm;
    ncol[j] = n;
    int ncl = (n < N) ? n : (N - 1);         // clamp row for safe loads, mask stores later
    brow[j] = W + (size_t)ncl * K + kk;
  }

  v8f acc0[4] = {v8f{}, v8f{}, v8f{}, v8f{}};   // M-subtile 0, N-subtiles 0..3
  v8f acc1[4] = {v8f{}, v8f{}, v8f{}, v8f{}};   // M-subtile 1

#pragma unroll 2
  for (int k0 = 0; k0 < K; k0 += 16) {
    v2f af0[4], af1[4], bf[4][4];
    // ---- load phase: 24 independent b64 loads (distinct registers) ----
#pragma unroll
    for (int u = 0; u < 4; ++u) {
      af0[u] = *(const v2f*)(arow0 + k0 + 4 * u);
      af1[u] = *(const v2f*)(arow1 + k0 + 4 * u);
#pragma unroll
      for (int j = 0; j < 4; ++j)
        bf[j][u] = *(const v2f*)(brow[j] + k0 + 4 * u);
    }
    // ---- compute phase: 32 WMMAs ----
#pragma unroll
    for (int u = 0; u < 4; ++u) {
#pragma unroll
      for (int j = 0; j < 4; ++j) {
        acc0[j] = __builtin_amdgcn_wmma_f32_16x16x4_f32(false, af0[u], false, bf[j][u],
                                                        (short)0, acc0[j], false, false);
        acc1[j] = __builtin_amdgcn_wmma_f32_16x16x4_f32(false, af1[u], false, bf[j][u],
                                                        (short)0, acc1[j], false, false);
      }
    }
  }

  // C/D layout: VGPR i -> row = tile + i + 8*(lane>=16), col = tileN + j*16 + (lane&15)
  const int rowOff = (lane >> 4) * 8;
#pragma unroll
  for (int j = 0; j < 4; ++j) {
    int n = ncol[j];
    if (n >= N) continue;
    float bv = bias ? bias[n] : 0.0f;
#pragma unroll
    for (int i = 0; i < 8; ++i) {
      float v0 = acc0[j][i] + bv;
      float v1 = acc1[j][i] + bv;
      if (act == 1) { v0 = fmaxf(v0, 0.0f); v1 = fmaxf(v1, 0.0f); }
      int row0 = tileM + rowOff + i;
      int row1 = tileM + 16 + rowOff + i;
      C[(size_t)row0 * ldc + n] = v0;
      C[(size_t)row1 * ldc + n] = v1;
      if (C2) {
        C2[(size_t)row0 * ldc2 + n] = v0;
        C2[(size_t)row1 * ldc2 + n] = v1;
      }
    }
  }
}

// ---------------------------------------------------------------------------
// avg[b,e] = mean_p image_feat[b,p,e]
// ---------------------------------------------------------------------------
__global__ __launch_bounds__(256) void avg_kernel(const float* __restrict__ feat,
                                                  float* __restrict__ avg)
{
  int b = blockIdx.x, tid = threadIdx.x;
  const float* fb = feat + (size_t)b * P_ * ENC_;
  for (int e = tid; e < ENC_; e += 256) {
    float s = 0.0f;
    for (int p = 0; p < P_; ++p) s += fb[(size_t)p * ENC_ + e];
    avg[b * ENC_ + e] = s * (1.0f / (float)P_);
  }
}

// ---------------------------------------------------------------------------
// Wcat[n, 0:1024] = W_ih[n,:], Wcat[n, 1024:1536] = W_hh[n,:]; bcat = b_ih + b_hh
// ---------------------------------------------------------------------------
__global__ __launch_bounds__(256) void build_wcat(
    const float* __restrict__ Wih, const float* __restrict__ Whh,
    const float* __restrict__ bih, const float* __restrict__ bhh,
    float* __restrict__ Wcat, float* __restrict__ bcat)
{
  int idx = blockIdx.x * 256 + threadIdx.x;
  const int total = NGATE_ * KCAT_;
  if (idx < total) {
    int n = idx / KCAT_;
    int k = idx - n * KCAT_;
    Wcat[idx] = (k < ENC_ + EMB_) ? Wih[(size_t)n * (ENC_ + EMB_) + k]
                                  : Whh[(size_t)n * DEC_ + (k - (ENC_ + EMB_))];
  }
  if (idx < NGATE_) bcat[idx] = bih[idx] + bhh[idx];
}

// ---------------------------------------------------------------------------
// Attention for one step: per-batch block (64 blocks x 256 thr = 8 waves).
//   e[p]   = sum_a tanh(enc_proj[b,p,a] + dec_proj[b,a]) * V[a] + Vb
//   w      = softmax_p(e)
//   ctx[e] = sum_p w[p] * feat[b,p,e]          -> xcat[b, 0:512]
//   emb    = embed_w[caps[b,t]]                -> xcat[b, 512:1024]
// ---------------------------------------------------------------------------
__global__ __launch_bounds__(256) void attention_kernel(
    const float* __restrict__ enc_proj, const float* __restrict__ dec_proj,
    const float* __restrict__ Vw, const float* __restrict__ Vb,
    const float* __restrict__ feat, const float* __restrict__ embed_w,
    const int* __restrict__ caps, float* __restrict__ xcat, int t)
{
  __shared__ __align__(16) float sdec[ATT_];
  __shared__ __align__(16) float sV[ATT_];
  __shared__ float se[224];
  __shared__ float sred[1];

  const int b = blockIdx.x;
  const int tid = threadIdx.x;
  for (int i = tid; i < ATT_; i += 256) {
    sdec[i] = dec_proj[b * ATT_ + i];
    sV[i]   = Vw[i];
  }
  __syncthreads();

  const int wave = tid >> 5, lane = tid & 31;
  const float* ep_b = enc_proj + (size_t)b * P_ * ATT_;
  const float vb = Vb[0];

  for (int p = wave; p < P_; p += 8) {
    const float* row = ep_b + (size_t)p * ATT_;
    float s = 0.0f;
#pragma unroll
    for (int i = 0; i < 4; ++i) {
      int a = lane * 4 + i * 128;
      float4 ev = *(const float4*)(row + a);
      float4 dv = *(const float4*)(sdec + a);
      float4 vv = *(const float4*)(sV + a);
      s += tanhf(ev.x + dv.x) * vv.x;
      s += tanhf(ev.y + dv.y) * vv.y;
      s += tanhf(ev.z + dv.z) * vv.z;
      s += tanhf(ev.w + dv.w) * vv.w;
    }
#pragma unroll
    for (int off = 16; off > 0; off >>= 1) s += __shfl_xor(s, off, 32);
    if (lane == 0) se[p] = s + vb;
  }
  __syncthreads();

  if (wave == 0) {                                // softmax stats on one wave32
    float m = -1e30f;
    for (int p = lane; p < P_; p += 32) m = fmaxf(m, se[p]);
#pragma unroll
    for (int off = 16; off > 0; off >>= 1) m = fmaxf(m, __shfl_xor(m, off, 32));
    float sum = 0.0f;
    for (int p = lane; p < P_; p += 32) { float ex = __expf(se[p] - m); se[p] = ex; sum += ex; }
#pragma unroll
    for (int off = 16; off > 0; off >>= 1) sum += __shfl_xor(sum, off, 32);
    if (lane == 0) sred[0] = 1.0f / sum;
  }
  __syncthreads();
  if (tid < P_) se[tid] *= sred[0];
  __syncthreads();

  const float* fb = feat + (size_t)b * P_ * ENC_;
  for (int e = tid; e < ENC_; e += 256) {
    float acc = 0.0f;
    for (int p = 0; p < P_; ++p) acc = fmaf(se[p], fb[(size_t)p * ENC_ + e], acc);
    xcat[(size_t)b * KCAT_ + e] = acc;
  }
  const int token = caps[b * S_ + t];
  const float* er = embed_w + (size_t)token * EMB_;
  for (int i = tid; i < EMB_; i += 256)
    xcat[(size_t)b * KCAT_ + ENC_ + i] = er[i];
}

// ---------------------------------------------------------------------------
// LSTM cell update. gates layout = [i | f | g | o], each DEC wide.
// Writes c, h, and mirrors h into xcat[b, 1024:1536] for the next step.
// ---------------------------------------------------------------------------
__global__ __launch_bounds__(256) void lstm_cell_kernel(
    const float* __restrict__ gates, float* __restrict__ cbuf,
    float* __restrict__ hbuf, float* __restrict__ xcat)
{
  int idx = blockIdx.x * 256 + threadIdx.x;      // 0 .. B*DEC-1
  int b = idx >> 9, d = idx & (DEC_ - 1);
  const float* g = gates + (size_t)b * NGATE_;
  float gi = g[d], gf = g[DEC_ + d], gg = g[2 * DEC_ + d], go = g[3 * DEC_ + d];
  float c  = cbuf[idx];
  float si = 1.0f / (1.0f + __expf(-gi));
  float sf = 1.0f / (1.0f + __expf(-gf));
  float so = 1.0f / (1.0f + __expf(-go));
  float cn = sf * c + si * tanhf(gg);
  float hn = so * tanhf(cn);
  cbuf[idx] = cn;
  hbuf[idx] = hn;
  xcat[(size_t)b * KCAT_ + ENC_ + EMB_ + d] = hn;
}

// ---------------------------------------------------------------------------
// Host-side orchestration (graph-capture safe: stream-ordered launches only).
// ---------------------------------------------------------------------------
extern "C" void kernel_launch(void* const* d_in, const int* in_sizes, int n_in,
                              void* d_out, int out_size, void* d_ws, size_t ws_size,
                              hipStream_t stream) {
  const float* image_feat = (const float*)d_in[0];   // [B,P,ENC]
  const int*   caps       = (const int*)  d_in[1];   // [B,S]
  const float* wenc_w     = (const float*)d_in[2];   // [ATT,ENC]
  const float* wenc_b     = (const float*)d_in[3];
  const float* wdec_w     = (const float*)d_in[4];   // [ATT,DEC]
  const float* wdec_b     = (const float*)d_in[5];
  const float* V_w        = (const float*)d_in[6];   // [1,ATT]
  const float* V_b        = (const float*)d_in[7];   // [1]
  const float* embed_w    = (const float*)d_in[8];   // [VOCAB,EMB]
  const float* h0_w       = (const float*)d_in[9];   // [DEC,ENC]
  const float* h0_b       = (const float*)d_in[10];
  const float* c0_w       = (const float*)d_in[11];
  const float* c0_b       = (const float*)d_in[12];
  const float* W_ih       = (const float*)d_in[13];  // [4*DEC, ENC+EMB]
  const float* b_ih       = (const float*)d_in[14];
  const float* W_hh       = (const float*)d_in[15];  // [4*DEC, DEC]
  const float* b_hh       = (const float*)d_in[16];
  const float* fc_w       = (const float*)d_in[17];  // [VOCAB,DEC]
  const float* fc_b       = (const float*)d_in[18];
  float* out = (float*)d_out;                        // [B,S,VOCAB]

  // Workspace layout (floats)
  float* ws = (float*)d_ws;
  float* enc_proj = ws;                                      // 12544*512
  float* Wcat     = enc_proj + (size_t)B_ * P_ * ATT_;       // 2048*1536
  float* bcat     = Wcat + (size_t)NGATE_ * KCAT_;           // 2048
  float* avg      = bcat + NGATE_;                           // 64*512
  float* hbuf     = avg  + B_ * ENC_;                        // 64*512
  float* cbuf     = hbuf + B_ * DEC_;                        // 64*512
  float* dec_proj = cbuf + B_ * DEC_;                        // 64*512
  float* xcat     = dec_proj + B_ * ATT_;                    // 64*1536
  float* gates    = xcat + (size_t)B_ * KCAT_;               // 64*2048

  // ---- step-invariant precompute ----
  avg_kernel<<<B_, 256, 0, stream>>>(image_feat, avg);
  build_wcat<<<(NGATE_ * KCAT_ + 255) / 256, 256, 0, stream>>>(W_ih, W_hh, b_ih, b_hh, Wcat, bcat);

  // enc_proj = image_feat @ wenc_w^T + wenc_b : M=12544, N=512, K=512
  // Mtiles = 12544/32 = 392, nstrips = 8 -> 3136 waves -> 392 blocks
  wmma_gemm_f32<<<392, 256, 0, stream>>>(image_feat, wenc_w, wenc_b, enc_proj, nullptr,
                                         B_ * P_, ATT_, ENC_, ATT_, 0, ATT_ / 64, 0);
  // h0 = relu(avg @ h0_w^T + b), mirrored into xcat h-slot; c0 likewise
  // Mtiles = 2, nstrips = 8 -> 16 waves -> 2 blocks
  wmma_gemm_f32<<<2, 256, 0, stream>>>(avg, h0_w, h0_b, hbuf, xcat + ENC_ + EMB_,
                                       B_, DEC_, ENC_, DEC_, KCAT_, DEC_ / 64, 1);
  wmma_gemm_f32<<<2, 256, 0, stream>>>(avg, c0_w, c0_b, cbuf, nullptr,
                                       B_, DEC_, ENC_, DEC_, 0, DEC_ / 64, 1);

  // ---- recurrent steps ----
  for (int t = 0; t < S_; ++t) {
    // dec_proj = h @ wdec_w^T + wdec_b : M=64, N=512, K=512 (2 blocks)
    wmma_gemm_f32<<<2, 256, 0, stream>>>(hbuf, wdec_w, wdec_b, dec_proj, nullptr,
                                         B_, ATT_, DEC_, ATT_, 0, ATT_ / 64, 0);
    // attention + context + embedding gather -> xcat[:, 0:1024]
    attention_kernel<<<B_, 256, 0, stream>>>(enc_proj, dec_proj, V_w, V_b,
                                             image_feat, embed_w, caps, xcat, t);
    // gates = xcat @ Wcat^T + bcat : M=64, N=2048, K=1536
    // Mtiles = 2, nstrips = 32 -> 64 waves -> 8 blocks
    wmma_gemm_f32<<<8, 256, 0, stream>>>(xcat, Wcat, bcat, gates, nullptr,
                                         B_, NGATE_, KCAT_, NGATE_, 0, NGATE_ / 64, 0);
    // LSTM cell -> h, c, and h mirrored into xcat
    lstm_cell_kernel<<<(B_ * DEC_) / 256, 256, 0, stream>>>(gates, cbuf, hbuf, xcat);
    // logits = h @ fc_w^T + fc_b, written strided into out[:, t, :]
    // M=64, N=10000, K=512, ldc = S*VOCAB; Mtiles=2 x 157 strips = 314 waves -> 40 blocks
    wmma_gemm_f32<<<40, 256, 0, stream>>>(hbuf, fc_w, fc_b, out + (size_t)t * VOCAB_, nullptr,
                                          B_, VOCAB_, DEC_, S_ * VOCAB_, 0, 157, 0);
  }
}